// LigerFusedLinearDPOLoss_57698590654887
// MI455X (gfx1250) — compile-verified
//
#include <hip/hip_runtime.h>
#include <math.h>
#include <stdint.h>

typedef __attribute__((ext_vector_type(2))) float v2f;
typedef __attribute__((ext_vector_type(8))) float v8f;

#define HD      4096          // hidden
#define VO      32000         // vocab
#define SEQS    8
#define TLEN    512
#define NTOK    (SEQS * TLEN) // 4096 tokens
#define IGNORE_IDX (-100)
#define BETA_F  0.1f

#define MT      16            // token rows per block
#define NWAVES  8             // waves per block
#define NSTRIDE (NWAVES * 16) // 128 vocab cols per macro step (32000/128 = 250 exact)
#define KC      256           // K-chunk per LDS buffer (double-buffered)
#define RSTR    (KC + 4)      // padded LDS row stride (floats); 260*4B = 65*16B -> b128 aligned,
                              // 4-float skew per row kills the 16-way bank conflict
#define ABUF    (MT * RSTR)           // 4160 floats
#define BBUF    (NSTRIDE * RSTR)      // 33280 floats
#define BUFS    (ABUF + BBUF)         // 37440 floats per pipeline stage
#define SHMEM_BYTES ((size_t)(2 * BUFS) * sizeof(float))  // 299,520 B dynamic LDS

// ---------------------------------------------------------------------------
// CDNA5 async global->LDS copy (ASYNCcnt-tracked, no VGPR round trip).
// VDST VGPR holds the workgroup-relative LDS byte offset.
// ---------------------------------------------------------------------------
__device__ __forceinline__ void async_copy_b128(uint32_t lds_off, const float* src) {
    asm volatile("global_load_async_to_lds_b128 %0, %1, off"
                 :: "v"(lds_off), "v"((uint64_t)(uintptr_t)src)
                 : "memory");
}
__device__ __forceinline__ void wait_async0() {
    asm volatile("s_wait_asynccnt 0x0" ::: "memory");
}

// ---------------------------------------------------------------------------
// Kernel 1: fused linear (f32 WMMA) + online log-softmax, with double-buffered
// async-LDS pipelining of the A/B K-chunks.
// grid.x = NTOK/MT m-tiles, grid.y = 2 (0: policy weights, 1: reference).
// ---------------------------------------------------------------------------
__global__ __launch_bounds__(256) void dpo_fused_logp_kernel(
    const float* __restrict__ W0, const float* __restrict__ b0,
    const float* __restrict__ W1, const float* __restrict__ b1,
    const float* __restrict__ X,  const int* __restrict__ tgt,
    float* __restrict__ tok_out)
{
    const int which = blockIdx.y;
    const float* __restrict__ W  = which ? W1 : W0;
    const float* __restrict__ Bv = which ? b1 : b0;
    const int mbase = blockIdx.x * MT;

    // Static LDS: softmax partial merge + target logits (16.4 KB)
    __shared__ float2 red[MT][NSTRIDE];
    __shared__ float  tl[MT];

    // Dynamic LDS: two pipeline stages of [A(16 x RSTR) | B(128 x RSTR)]
    extern __shared__ __align__(16) float lds[];
    // Byte offset of the dynamic LDS region within the workgroup allocation —
    // this is what the async VDST operand needs (no addrspacecast involved).
    const uint32_t dyn0 = __builtin_amdgcn_groupstaticsize();

    const int tid   = threadIdx.x;
    const int wave  = tid >> 5;
    const int lane  = tid & 31;
    const int hlane = lane & 15;
    const int hi    = lane >> 4;     // lane half: K-offset 0/2 (A,B frags), M-offset 0/8 (C frag)

    if (tid < MT) tl[tid] = 0.0f;

    // Issue async staging of chunk (nb_, kc_) into pipeline stage b (0/1).
    auto stage_chunk = [&](int nb_, int kc_, int b) {
        const uint32_t abase = dyn0 + (uint32_t)(b * BUFS) * 4u;
        const uint32_t bbase = abase + (uint32_t)ABUF * 4u;
#pragma unroll
        for (int j = 0; j < (MT * (KC / 4)) / 256; ++j) {        // A: 4 issues/thread
            const int i = j * 256 + tid;
            const int row = i >> 6, kq = i & 63;                 // 64 float4 per row
            async_copy_b128(abase + (uint32_t)(row * RSTR + kq * 4) * 4u,
                            X + (size_t)(mbase + row) * HD + kc_ + kq * 4);
        }
#pragma unroll
        for (int j = 0; j < (NSTRIDE * (KC / 4)) / 256; ++j) {   // B: 32 issues/thread
            const int i = j * 256 + tid;
            const int row = i >> 6, kq = i & 63;
            async_copy_b128(bbase + (uint32_t)(row * RSTR + kq * 4) * 4u,
                            W + (size_t)(nb_ + row) * HD + kc_ + kq * 4);
        }
    };

    // Targets for the 8 C-matrix rows this lane's fragment covers.
    int trow[8], tg[8];
#pragma unroll
    for (int r = 0; r < 8; ++r) {
        trow[r] = hi ? (r + 8) : r;
        tg[r]   = tgt[mbase + trow[r]];
    }

    float m_acc[8], s_acc[8], t_val[8];
    unsigned found = 0u;
#pragma unroll
    for (int r = 0; r < 8; ++r) { m_acc[r] = -__builtin_inff(); s_acc[r] = 0.0f; t_val[r] = 0.0f; }

    for (int nb = 0; nb < VO; nb += NSTRIDE) {
        const int   col    = nb + wave * 16 + hlane;   // this lane's B/C column
        const float bias_c = Bv[col];

        v8f c = {};
        stage_chunk(nb, 0, 0);            // prologue fill of stage 0
        wait_async0();
        __syncthreads();

        int cur = 0;
        for (int kc = 0; kc < HD; kc += KC) {
            // Kick off the DMA of the next chunk; it rides under the WMMA loop.
            if (kc + KC < HD) stage_chunk(nb, kc + KC, cur ^ 1);

            const float* base = lds + cur * BUFS;      // runtime-indexed, no const-aggregate
            const float* ap = base + hlane * RSTR + 2 * hi;
            const float* bp = base + ABUF + (wave * 16 + hlane) * RSTR + 2 * hi;
#pragma unroll 8
            for (int k = 0; k < KC; k += 4) {
                const float2 av = *(const float2*)(ap + k);
                const float2 bv = *(const float2*)(bp + k);
                v2f a, b;
                a.x = av.x; a.y = av.y;
                b.x = bv.x; b.y = bv.y;
                c = __builtin_amdgcn_wmma_f32_16x16x4_f32(
                        false, a, false, b, (short)0, c, false, false);
            }
            wait_async0();                // my async writes for cur^1 have landed
            __syncthreads();              // everyone done reading cur / writing cur^1
            cur ^= 1;
        }

        // ---- streaming log-softmax update + target logit capture ----
#pragma unroll
        for (int r = 0; r < 8; ++r) {
            float logit = c[r] + bias_c;
            if (tg[r] == col) { t_val[r] = logit; found |= (1u << r); }
            float nm = fmaxf(m_acc[r], logit);
            s_acc[r] = s_acc[r] * __expf(m_acc[r] - nm) + __expf(logit - nm);
            m_acc[r] = nm;
        }
    }

    // Deposit per-lane streaming-softmax partials + target logits.
#pragma unroll
    for (int r = 0; r < 8; ++r) {
        float2 p; p.x = m_acc[r]; p.y = s_acc[r];
        red[trow[r]][wave * 16 + hlane] = p;
        if (found & (1u << r)) tl[trow[r]] = t_val[r];
    }
    __syncthreads();

    // One thread per row: merge 128 (max,sumexp) partials, emit masked token logp.
    if (tid < MT) {
        float M = -__builtin_inff(), S = 0.0f;
        for (int j = 0; j < NSTRIDE; ++j) {
            float2 p = red[tid][j];
            float nm = fmaxf(M, p.x);
            S = S * __expf(M - nm) + p.y * __expf(p.x - nm);
            M = nm;
        }
        const int row_g = mbase + tid;
        const int t     = tgt[row_g];
        const float logZ = M + __logf(S);
        tok_out[which * NTOK + row_g] = (t == IGNORE_IDX) ? 0.0f : (tl[tid] - logZ);
    }
}

// ---------------------------------------------------------------------------
// Kernel 2: per-sequence sums (deterministic wave reductions) + DPO epilogue.
// ---------------------------------------------------------------------------
__global__ __launch_bounds__(256) void dpo_final_kernel(
    const float* __restrict__ tok, const int* __restrict__ tgt,
    float* __restrict__ out)
{
    __shared__ float sp[SEQS], sr[SEQS], mc[SEQS];
    const int tid  = threadIdx.x;
    const int wave = tid >> 5;   // wave i owns sequence i (8 waves exactly)
    const int lane = tid & 31;

    float accp = 0.0f, accr = 0.0f, accm = 0.0f;
    for (int i = lane; i < TLEN; i += 32) {
        const int idx = wave * TLEN + i;
        accp += tok[idx];
        accr += tok[NTOK + idx];
        accm += (tgt[idx] != IGNORE_IDX) ? 1.0f : 0.0f;
    }
#pragma unroll
    for (int off = 16; off > 0; off >>= 1) {
        accp += __shfl_down(accp, off, 32);
        accr += __shfl_down(accr, off, 32);
        accm += __shfl_down(accm, off, 32);
    }
    if (lane == 0) { sp[wave] = accp; sr[wave] = accr; mc[wave] = accm; }
    __syncthreads();

    if (tid == 0) {
        float nll_num = 0.0f, mask = 0.0f;
        for (int s = 0; s < 4; ++s) { nll_num += sp[s]; mask += mc[s]; }
        const float nll = -nll_num / fmaxf(mask, 1.0f);

        float dpo = 0.0f;
        for (int i = 0; i < 4; ++i) {
            const float diff = BETA_F * ((sp[i] - sr[i]) - (sp[i + 4] - sr[i + 4]));
            const float y = -diff;   // -log_sigmoid(diff) = softplus(-diff)
            dpo += (y > 0.0f) ? (y + log1pf(__expf(-y))) : log1pf(__expf(y));
        }
        out[0] = nll + dpo * 0.25f;
    }
}

// ---------------------------------------------------------------------------
extern "C" void kernel_launch(void* const* d_in, const int* in_sizes, int n_in,
                              void* d_out, int out_size, void* d_ws, size_t ws_size,
                              hipStream_t stream) {
    const float* lin_w  = (const float*)d_in[0];
    const float* x      = (const float*)d_in[1];
    const int*   target = (const int*)  d_in[2];
    const float* bias   = (const float*)d_in[3];
    const float* ref_w  = (const float*)d_in[4];
    const float* ref_b  = (const float*)d_in[5];

    float* tok_ws = (float*)d_ws;   // [2][NTOK] masked token log-probs

    dim3 grid(NTOK / MT, 2);
    dpo_fused_logp_kernel<<<grid, 256, SHMEM_BYTES, stream>>>(
        lin_w, bias, ref_w, ref_b, x, target, tok_ws);

    dpo_final_kernel<<<1, 256, 0, stream>>>(tok_ws, target, (float*)d_out);
}